// MVMD_21766894256685
// MI455X (gfx1250) — compile-verified
//
#include <hip/hip_runtime.h>
#include <math.h>

// ---------------------------------------------------------------------------
// MVMD on MI455X (gfx1250).
//  - Iteration core: elementwise sweep kernels (L2-resident state, wave32).
//  - All FFTs (F=16384 fwd, F=16384 inv, T=8192 fwd) as four-step FFTs whose
//    128/64-point DFT stages are fp32 complex matmuls on V_WMMA_F32_16X16X4_F32.
// Sizes: C=16, T=8192, F=16384, H=8192, K=8, ALPHA=2000, MAX_N=20 (19 sweeps).
// ---------------------------------------------------------------------------

typedef __attribute__((ext_vector_type(2))) float v2f;
typedef __attribute__((ext_vector_type(8))) float v8f;

#define TPI 6.283185307179586f

// ---- workspace layout (float offsets) ----
constexpr size_t OFF_D128R = 0;                         // 128x128 DFT matrix, real
constexpr size_t OFF_D128I = OFF_D128R + 16384;         // imag
constexpr size_t OFF_D64R  = OFF_D128I + 16384;         // 64x64 DFT matrix
constexpr size_t OFF_D64I  = OFF_D64R  + 4096;
constexpr size_t OFF_OMEGA = OFF_D64I  + 4096;          // omega_cur[8]
constexpr size_t OFF_FPOSR = OFF_OMEGA + 16;            // fpos (H x C) split planes
constexpr size_t OFF_FPOSI = OFF_FPOSR + 131072;
constexpr size_t OFF_UR    = OFF_FPOSI + 131072;        // u[k] planes (8 x H x C)
constexpr size_t OFF_UI    = OFF_UR    + 1048576;
constexpr size_t OFF_SUMR  = OFF_UI    + 1048576;       // sum_uk (H x C)
constexpr size_t OFF_SUMI  = OFF_SUMR  + 131072;
constexpr size_t OFF_PART  = OFF_SUMI  + 131072;        // reduction partials 512x8x2
constexpr size_t OFF_ZR    = OFF_PART  + 8192;          // shared FFT scratch planes
constexpr size_t OFF_ZI    = OFF_ZR    + 2097152;       // (max 128 transforms x 16384)
constexpr size_t OFF_SMAT  = OFF_ZI    + 2097152;       // time-domain slices 128x8192

// ---- output layout (float offsets into d_out) ----
constexpr size_t OUT_UOUT  = 0;                          // (K,T,C) float32
constexpr size_t OUT_UHAT2 = 1048576;                    // (T,K,C) complex64
constexpr size_t OUT_OMEGA = OUT_UHAT2 + 2097152;        // (20,K) complex64

// ---------------------------------------------------------------------------
// WMMA helpers: V_WMMA_F32_16X16X4_F32, D = A(16x4) * B(4x16) + C(16x16).
// Fragment layouts per CDNA5 ISA 7.12.2 (fp32, wave32).
// ---------------------------------------------------------------------------
__device__ __forceinline__ v8f wmma4(v2f a, v2f b, v8f c) {
  return __builtin_amdgcn_wmma_f32_16x16x4_f32(false, a, false, b, (short)0, c,
                                               false, false);
}
// A fragment: element (m0+ (lane&15), k0 + 2*(lane>>4) + v) from row-major P[ld]
__device__ __forceinline__ v2f ldA(const float* P, int ld, int m0, int k0, int lane) {
  const float* q = P + (size_t)(m0 + (lane & 15)) * ld + k0 + ((lane >> 4) << 1);
  v2f r; r.x = q[0]; r.y = q[1]; return r;
}
// B fragment with transposed source: B(k,n) = P[n*ld + k]  (for B = D^T)
__device__ __forceinline__ v2f ldBt(const float* P, int ld, int k0, int n0, int lane) {
  const float* q = P + (size_t)(n0 + (lane & 15)) * ld + k0 + ((lane >> 4) << 1);
  v2f r; r.x = q[0]; r.y = q[1]; return r;
}

__device__ __forceinline__ float mirror_sig(const float* sig, int ch, int n) {
  // f = [sig[:,:4096][::-1], sig, sig[:,4096:][::-1]]
  int idx = (n < 4096) ? (4095 - n) : ((n < 12288) ? (n - 4096) : (20479 - n));
  return sig[ch * 8192 + idx];
}

// Hermitian gather of full inverse-FFT input U[m] from stored positive half.
__device__ __forceinline__ void loadU(const float* ur, const float* ui, int mode,
                                      int ch, int m, float& re, float& im) {
  int j = (m < 8192) ? m : ((m == 8192) ? 8191 : (16384 - m));
  float sg = (m < 8192) ? 1.0f : -1.0f;
  size_t o = (size_t)mode * 131072 + (size_t)j * 16 + ch;
  re = ur[o]; im = sg * ui[o];
}

// ---------------------------------------------------------------------------
// Init kernels
// ---------------------------------------------------------------------------
__global__ void k_tables(float* ws, float* out) {
  int t = blockIdx.x * 256 + threadIdx.x;            // 0..16383
  int c = t >> 7, a = t & 127;
  int p = (a * c) & 127;
  float sn, cs;
  __sincosf(-TPI * (float)p * (1.0f / 128.0f), &sn, &cs);
  ws[OFF_D128R + t] = cs;
  ws[OFF_D128I + t] = sn;
  if (t < 4096) {
    int c6 = t >> 6, a6 = t & 63;
    int p6 = (a6 * c6) & 63;
    __sincosf(-TPI * (float)p6 * (1.0f / 64.0f), &sn, &cs);
    ws[OFF_D64R + t] = cs;
    ws[OFF_D64I + t] = sn;
  }
  if (t < 8) {                                        // omega0[k] = 0.5/K * k
    float o0 = (float)t * (1.0f / 16.0f);
    ws[OFF_OMEGA + t] = o0;
    out[OUT_OMEGA + 2 * t] = o0;
    out[OUT_OMEGA + 2 * t + 1] = 0.0f;
  }
}

__global__ void k_zero(float* ws) {
  const size_t n = 2359296;                           // u planes + sum planes
  for (size_t x = (size_t)blockIdx.x * 256 + threadIdx.x; x < n;
       x += (size_t)gridDim.x * 256)
    ws[OFF_UR + x] = 0.0f;
}

// ---------------------------------------------------------------------------
// Forward FFT, F=16384, per channel (real input, mirrored on the fly).
// Step1: Z = twiddle .* (D128 * X), X[a][b] = f[128a+b]   (real -> 2 wmmas/k)
// ---------------------------------------------------------------------------
__global__ void k_fft1_s1(const float* __restrict__ sig, float* ws) {
  int lane = threadIdx.x & 31;
  int w = blockIdx.x * 8 + (threadIdx.x >> 5);        // 1024 tile jobs
  int ch = w >> 6, rem = w & 63;
  int c0 = (rem >> 3) << 4, b0 = (rem & 7) << 4;
  const float* Dr = ws + OFF_D128R; const float* Di = ws + OFF_D128I;
  v8f yr = {0,0,0,0,0,0,0,0}, yi = {0,0,0,0,0,0,0,0};
  for (int kk = 0; kk < 32; kk++) {
    int k0 = kk << 2;
    v2f ar = ldA(Dr, 128, c0, k0, lane);
    v2f ai = ldA(Di, 128, c0, k0, lane);
    int aa = k0 + ((lane >> 4) << 1), bb = b0 + (lane & 15);
    float x0 = mirror_sig(sig, ch, 128 * aa + bb);
    float x1 = mirror_sig(sig, ch, 128 * aa + 128 + bb);
    v2f bx = {x0, x1};
    yr = wmma4(ar, bx, yr);
    yi = wmma4(ai, bx, yi);
  }
  float* Zr = ws + OFF_ZR + (size_t)ch * 16384;
  float* Zi = ws + OFF_ZI + (size_t)ch * 16384;
#pragma unroll
  for (int r = 0; r < 8; r++) {
    int c = c0 + r + ((lane >> 4) << 3), b = b0 + (lane & 15);
    float sn, cs;
    __sincosf(-TPI * (float)(c * b) * (1.0f / 16384.0f), &sn, &cs);
    float vr = yr[r], vi = yi[r];
    Zr[c * 128 + b] = vr * cs - vi * sn;
    Zi[c * 128 + b] = vr * sn + vi * cs;
  }
}

// Step3: fpos[i=128d+c][ch] = sum_b Z[c][b]*D128[d][b], only d<64 (bins < H)
__global__ void k_fft1_s3(float* ws) {
  int lane = threadIdx.x & 31;
  int w = blockIdx.x * 8 + (threadIdx.x >> 5);        // 512 tile jobs
  int ch = w >> 5, rem = w & 31;
  int c0 = (rem >> 2) << 4, d0 = (rem & 3) << 4;
  const float* Dr = ws + OFF_D128R; const float* Di = ws + OFF_D128I;
  const float* Zr = ws + OFF_ZR + (size_t)ch * 16384;
  const float* Zi = ws + OFF_ZI + (size_t)ch * 16384;
  v8f orr = {0,0,0,0,0,0,0,0}, oii = {0,0,0,0,0,0,0,0};
  for (int kk = 0; kk < 32; kk++) {
    int k0 = kk << 2;
    v2f zr = ldA(Zr, 128, c0, k0, lane);
    v2f zi = ldA(Zi, 128, c0, k0, lane);
    v2f br = ldBt(Dr, 128, k0, d0, lane);
    v2f bi = ldBt(Di, 128, k0, d0, lane);
    v2f nz = -zi;
    orr = wmma4(zr, br, orr); orr = wmma4(nz, bi, orr);
    oii = wmma4(zr, bi, oii); oii = wmma4(zi, br, oii);
  }
#pragma unroll
  for (int r = 0; r < 8; r++) {
    int c = c0 + r + ((lane >> 4) << 3), d = d0 + (lane & 15);
    size_t i = (size_t)(128 * d + c) * 16 + ch;
    ws[OFF_FPOSR + i] = orr[r];
    ws[OFF_FPOSI + i] = oii[r];
  }
}

// ---------------------------------------------------------------------------
// One ADMM sweep: per-element mode chain + block partials for omega update.
// ---------------------------------------------------------------------------
__global__ void k_update(float* ws) {
  int e = blockIdx.x * 256 + threadIdx.x;             // 0..131071
  int i = e >> 4;
  float fr = (float)i * (1.0f / 16384.0f);            // freqs[F/2+i] == fh[i]
  float pr = ws[OFF_FPOSR + e], pi = ws[OFF_FPOSI + e];
  float sr = ws[OFF_SUMR + e], si = ws[OFF_SUMI + e];
  float uor[8], uoi[8], nwr[8], nwi[8], wsum[8], psum[8];
#pragma unroll
  for (int k = 0; k < 8; k++) {
    uor[k] = ws[OFF_UR + (size_t)k * 131072 + e];
    uoi[k] = ws[OFF_UI + (size_t)k * 131072 + e];
  }
  sr += uor[7] - uor[0]; si += uoi[7] - uoi[0];
#pragma unroll
  for (int k = 0; k < 8; k++) {
    if (k > 0) { sr += nwr[k - 1] - uor[k]; si += nwi[k - 1] - uoi[k]; }
    float dwf = fr - ws[OFF_OMEGA + k];               // uniform scalar load
    float inv = 1.0f / (1.0f + 2000.0f * dwf * dwf);  // ALPHA = 2000
    nwr[k] = (pr - sr) * inv;
    nwi[k] = (pi - si) * inv;
    float mag = nwr[k] * nwr[k] + nwi[k] * nwi[k];
    wsum[k] = fr * mag;
    psum[k] = mag;
  }
#pragma unroll
  for (int k = 0; k < 8; k++) {
    ws[OFF_UR + (size_t)k * 131072 + e] = nwr[k];
    ws[OFF_UI + (size_t)k * 131072 + e] = nwi[k];
  }
  ws[OFF_SUMR + e] = sr; ws[OFF_SUMI + e] = si;

  __shared__ float red[256];
  for (int q = 0; q < 16; q++) {
    int k = q >> 1, isP = q & 1;
    red[threadIdx.x] = isP ? psum[k] : wsum[k];
    __syncthreads();
    for (int s = 128; s > 0; s >>= 1) {
      if (threadIdx.x < s) red[threadIdx.x] += red[threadIdx.x + s];
      __syncthreads();
    }
    if (threadIdx.x == 0)
      ws[OFF_PART + (size_t)(k * 512 + blockIdx.x) * 2 + isP] = red[0];
    __syncthreads();
  }
}

// Deterministic finalize: omega[k] = (sum fh*|u|^2)/(sum |u|^2); record history.
__global__ void k_omega(float* ws, float* out, int row) {
  __shared__ float rw[256], rp[256];
  for (int k = 0; k < 8; k++) {
    float w = 0.0f, p = 0.0f;
    for (int b = threadIdx.x; b < 512; b += 256) {
      w += ws[OFF_PART + (size_t)(k * 512 + b) * 2];
      p += ws[OFF_PART + (size_t)(k * 512 + b) * 2 + 1];
    }
    rw[threadIdx.x] = w; rp[threadIdx.x] = p;
    __syncthreads();
    for (int s = 128; s > 0; s >>= 1) {
      if (threadIdx.x < s) { rw[threadIdx.x] += rw[threadIdx.x + s];
                             rp[threadIdx.x] += rp[threadIdx.x + s]; }
      __syncthreads();
    }
    if (threadIdx.x == 0) {
      float o = rw[0] / rp[0];
      ws[OFF_OMEGA + k] = o;
      out[OUT_OMEGA + (size_t)(row * 8 + k) * 2] = o;
      out[OUT_OMEGA + (size_t)(row * 8 + k) * 2 + 1] = 0.0f;
    }
    __syncthreads();
  }
}

// ---------------------------------------------------------------------------
// Inverse FFT, F=16384, per (mode,chan); Hermitian U gathered inline.
// Step1: Z = twiddle+ .* (conj(D128) * U)
// ---------------------------------------------------------------------------
__global__ void k_ifft_s1(float* ws) {
  int lane = threadIdx.x & 31;
  int w = blockIdx.x * 8 + (threadIdx.x >> 5);        // 8192 tile jobs
  int t = w >> 6, rem = w & 63;
  int mode = t >> 4, ch = t & 15;
  int c0 = (rem >> 3) << 4, b0 = (rem & 7) << 4;
  const float* Dr = ws + OFF_D128R; const float* Di = ws + OFF_D128I;
  const float* ur = ws + OFF_UR;   const float* ui = ws + OFF_UI;
  v8f yr = {0,0,0,0,0,0,0,0}, yi = {0,0,0,0,0,0,0,0};
  for (int kk = 0; kk < 32; kk++) {
    int k0 = kk << 2;
    v2f dr = ldA(Dr, 128, c0, k0, lane);
    v2f di = ldA(Di, 128, c0, k0, lane);
    int aa = k0 + ((lane >> 4) << 1), bb = b0 + (lane & 15);
    float r0, i0, r1, i1;
    loadU(ur, ui, mode, ch, 128 * aa + bb, r0, i0);
    loadU(ur, ui, mode, ch, 128 * aa + 128 + bb, r1, i1);
    v2f br = {r0, r1};
    v2f bi = {i0, i1};
    v2f nd = -di;
    // conj(D)*U: Re += Dr*Ur + Di*Ui ; Im += Dr*Ui - Di*Ur
    yr = wmma4(dr, br, yr); yr = wmma4(di, bi, yr);
    yi = wmma4(dr, bi, yi); yi = wmma4(nd, br, yi);
  }
  float* Zr = ws + OFF_ZR + (size_t)t * 16384;
  float* Zi = ws + OFF_ZI + (size_t)t * 16384;
#pragma unroll
  for (int r = 0; r < 8; r++) {
    int c = c0 + r + ((lane >> 4) << 3), b = b0 + (lane & 15);
    float sn, cs;
    __sincosf(TPI * (float)(c * b) * (1.0f / 16384.0f), &sn, &cs);
    float vr = yr[r], vi = yi[r];
    Zr[c * 128 + b] = vr * cs - vi * sn;
    Zi[c * 128 + b] = vr * sn + vi * cs;
  }
}

// Step3: only real part, only time window n in [4096,12288) (d-tiles 32..95).
// Re(Z * conj(D)^T) = Zr*Dr + Zi*Di  -> 2 wmmas per k-step.
__global__ void k_ifft_s3(float* ws, float* out) {
  int lane = threadIdx.x & 31;
  int w = blockIdx.x * 8 + (threadIdx.x >> 5);        // 4096 tile jobs
  int t = w >> 5, rem = w & 31;
  int mode = t >> 4, ch = t & 15;
  int c0 = (rem >> 2) << 4, d0 = 32 + ((rem & 3) << 4);
  const float* Dr = ws + OFF_D128R; const float* Di = ws + OFF_D128I;
  const float* Zr = ws + OFF_ZR + (size_t)t * 16384;
  const float* Zi = ws + OFF_ZI + (size_t)t * 16384;
  v8f o = {0,0,0,0,0,0,0,0};
  for (int kk = 0; kk < 32; kk++) {
    int k0 = kk << 2;
    v2f zr = ldA(Zr, 128, c0, k0, lane);
    v2f zi = ldA(Zi, 128, c0, k0, lane);
    v2f br = ldBt(Dr, 128, k0, d0, lane);
    v2f bi = ldBt(Di, 128, k0, d0, lane);
    o = wmma4(zr, br, o);
    o = wmma4(zi, bi, o);
  }
  int cout = (ch + 8) & 15;                           // ifftshift over channels
#pragma unroll
  for (int r = 0; r < 8; r++) {
    int c = c0 + r + ((lane >> 4) << 3), d = d0 + (lane & 15);
    int tt = 128 * d + c - 4096;                      // slice [F/4, 3F/4)
    float val = o[r] * (1.0f / 16384.0f);             // ifft scale
    ws[OFF_SMAT + (size_t)t * 8192 + tt] = val;
    out[OUT_UOUT + ((size_t)mode * 8192 + tt) * 16 + cout] = val;
  }
}

// ---------------------------------------------------------------------------
// Forward FFT, T=8192 = 128x64, per (mode,chan), real input from smat.
// ---------------------------------------------------------------------------
__global__ void k_fft2_s1(float* ws) {
  int lane = threadIdx.x & 31;
  int w = blockIdx.x * 8 + (threadIdx.x >> 5);        // 4096 tile jobs
  int t = w >> 5, rem = w & 31;
  int c0 = (rem >> 2) << 4, b0 = (rem & 3) << 4;
  const float* Dr = ws + OFF_D128R; const float* Di = ws + OFF_D128I;
  const float* s = ws + OFF_SMAT + (size_t)t * 8192;
  v8f yr = {0,0,0,0,0,0,0,0}, yi = {0,0,0,0,0,0,0,0};
  for (int kk = 0; kk < 32; kk++) {
    int k0 = kk << 2;
    v2f dr = ldA(Dr, 128, c0, k0, lane);
    v2f di = ldA(Di, 128, c0, k0, lane);
    int aa = k0 + ((lane >> 4) << 1), bb = b0 + (lane & 15);
    float x0 = s[64 * aa + bb];
    float x1 = s[64 * aa + 64 + bb];
    v2f bx = {x0, x1};
    yr = wmma4(dr, bx, yr);
    yi = wmma4(di, bx, yi);
  }
  float* Zr = ws + OFF_ZR + (size_t)t * 8192;
  float* Zi = ws + OFF_ZI + (size_t)t * 8192;
#pragma unroll
  for (int r = 0; r < 8; r++) {
    int c = c0 + r + ((lane >> 4) << 3), b = b0 + (lane & 15);
    float sn, cs;
    __sincosf(-TPI * (float)(c * b) * (1.0f / 8192.0f), &sn, &cs);
    float vr = yr[r], vi = yi[r];
    Zr[c * 64 + b] = vr * cs - vi * sn;
    Zi[c * 64 + b] = vr * sn + vi * cs;
  }
}

// Step3: X2[128d+c] = sum_b Z[c][b]*D64[d][b]; emit conj at fftshifted index.
__global__ void k_fft2_s3(float* ws, float* out) {
  int lane = threadIdx.x & 31;
  int w = blockIdx.x * 8 + (threadIdx.x >> 5);        // 4096 tile jobs
  int t = w >> 5, rem = w & 31;
  int mode = t >> 4, ch = t & 15;
  int c0 = (rem >> 2) << 4, d0 = (rem & 3) << 4;
  const float* Dr = ws + OFF_D64R; const float* Di = ws + OFF_D64I;
  const float* Zr = ws + OFF_ZR + (size_t)t * 8192;
  const float* Zi = ws + OFF_ZI + (size_t)t * 8192;
  v8f orr = {0,0,0,0,0,0,0,0}, oii = {0,0,0,0,0,0,0,0};
  for (int kk = 0; kk < 16; kk++) {
    int k0 = kk << 2;
    v2f zr = ldA(Zr, 64, c0, k0, lane);
    v2f zi = ldA(Zi, 64, c0, k0, lane);
    v2f br = ldBt(Dr, 64, k0, d0, lane);
    v2f bi = ldBt(Di, 64, k0, d0, lane);
    v2f nz = -zi;
    orr = wmma4(zr, br, orr); orr = wmma4(nz, bi, orr);
    oii = wmma4(zr, bi, oii); oii = wmma4(zi, br, oii);
  }
#pragma unroll
  for (int r = 0; r < 8; r++) {
    int c = c0 + r + ((lane >> 4) << 3), d = d0 + (lane & 15);
    int m = 128 * d + c;
    int tsh = (m + 4096) & 8191;                      // inverse of fftshift
    size_t ob = OUT_UHAT2 + (((size_t)tsh * 8 + mode) * 16 + ch) * 2;
    out[ob] = orr[r];
    out[ob + 1] = -oii[r];                            // conj
  }
}

// ---------------------------------------------------------------------------
extern "C" void kernel_launch(void* const* d_in, const int* in_sizes, int n_in,
                              void* d_out, int out_size, void* d_ws,
                              size_t ws_size, hipStream_t stream) {
  const float* sig = (const float*)d_in[0];           // (16, 8192) f32
  float* out = (float*)d_out;
  float* ws = (float*)d_ws;

  k_tables<<<64, 256, 0, stream>>>(ws, out);
  k_zero<<<1024, 256, 0, stream>>>(ws);

  // mirror + forward FFT(16384) per channel -> fpos (positive half only)
  k_fft1_s1<<<128, 256, 0, stream>>>(sig, ws);
  k_fft1_s3<<<64, 256, 0, stream>>>(ws);

  // 19 sequential VMD sweeps
  for (int n = 0; n < 19; n++) {
    k_update<<<512, 256, 0, stream>>>(ws);
    k_omega<<<1, 256, 0, stream>>>(ws, out, n + 1);
  }

  // Hermitian inverse FFT(16384) per (mode,chan) -> u_out + time slices
  k_ifft_s1<<<1024, 256, 0, stream>>>(ws);
  k_ifft_s3<<<512, 256, 0, stream>>>(ws, out);

  // forward FFT(8192) per (mode,chan) -> u_hat2 (conj, fftshifted)
  k_fft2_s1<<<512, 256, 0, stream>>>(ws);
  k_fft2_s3<<<512, 256, 0, stream>>>(ws, out);
}